// ExperimentalMSELoss_17935783428185
// MI455X (gfx1250) — compile-verified
//
#include <hip/hip_runtime.h>
#include <stdint.h>

// ---------------------------------------------------------------------------
// Problem geometry: pred/target are (64, 1, 512, 512) float32.
//   per-batch elements = 262144, batches = 64, total = 16.7M per input.
// Memory-bound: 128 MB read @ 23.3 TB/s => ~5.5us floor. Everything else
// (per-batch max/sum/hist + global weighted-SE sum) rides along in registers.
// ---------------------------------------------------------------------------

#define TPB          256
#define NBATCH       64
#define HW           262144            // 1*512*512 elements per batch
#define BPB          32                // blocks per batch
#define F4_PER_BLK   2048              // (HW/4)/BPB float4 per block
#define TILES        8                 // F4_PER_BLK / TPB  -> 1 float4/thread/tile
#define THRESH       0.1f

// Workspace layout (32-bit words), zeroed each launch:
//   [0,64)      loss[b]       (float, atomicAdd)
//   [64,128)    sum_pred[b]   (float)
//   [128,192)   sum_targ[b]   (float)
//   [192,256)   max_pred[b]   (uint bits, atomicMax; valid since values >= 0)
//   [256,320)   max_targ[b]   (uint bits)
//   [320,960)   hist_pred[b*10+i] (uint counts)
//   [960,1600)  hist_targ[b*10+i]
#define WS_WORDS 1600

typedef __attribute__((ext_vector_type(2))) float v2f;
typedef __attribute__((ext_vector_type(8))) float v8f;

// Async global->LDS 128-bit copy (CDNA5 async path, tracked by ASYNCcnt).
__device__ __forceinline__ void async_b128(unsigned lds_off, unsigned gl_off,
                                           const float* __restrict__ base) {
    asm volatile("global_load_async_to_lds_b128 %0, %1, %2 offset:0"
                 :: "v"(lds_off), "v"(gl_off), "s"(base)
                 : "memory");
}

__global__ __launch_bounds__(TPB)
void stats_kernel(const float* __restrict__ pred,
                  const float* __restrict__ targ,
                  float* __restrict__ ws) {
    // double buffer: [buf][input][tid] float4 = 16 KB LDS
    __shared__ float4 lbuf[2][2][TPB];
    __shared__ unsigned bhist[20];

    const int tid = threadIdx.x;
    const int b   = blockIdx.x / BPB;
    const int blk = blockIdx.x % BPB;
    if (tid < 20) bhist[tid] = 0u;

    const size_t base_e = (size_t)b * HW + (size_t)blk * (F4_PER_BLK * 4);
    const float* pbase = pred + base_e;
    const float* tbase = targ + base_e;

    float lossA = 0.f, sumP = 0.f, sumT = 0.f, maxP = 0.f, maxT = 0.f;
    unsigned long long packP = 0ull, packT = 0ull;   // 10 bins x 6 bits

    // --- prologue: prefetch tiles 0 and 1 (4 async ops in flight) ---
    {
        unsigned g0 = (unsigned)((0 * TPB + tid) * 16);
        unsigned g1 = (unsigned)((1 * TPB + tid) * 16);
        async_b128((unsigned)(uintptr_t)&lbuf[0][0][tid], g0, pbase);
        async_b128((unsigned)(uintptr_t)&lbuf[0][1][tid], g0, tbase);
        async_b128((unsigned)(uintptr_t)&lbuf[1][0][tid], g1, pbase);
        async_b128((unsigned)(uintptr_t)&lbuf[1][1][tid], g1, tbase);
    }

#pragma unroll
    for (int it = 0; it < TILES; ++it) {
        // async loads complete in order: <=2 outstanding means tile `it` landed
        if (it < TILES - 1)
            asm volatile("s_wait_asynccnt 0x2" ::: "memory");
        else
            asm volatile("s_wait_asynccnt 0x0" ::: "memory");

        const int cur = it & 1;
        float4 p4 = lbuf[cur][0][tid];
        float4 t4 = lbuf[cur][1][tid];

        // make sure the ds_loads have consumed the buffer before overwriting it
        asm volatile("s_wait_dscnt 0x0" ::: "memory");
        if (it + 2 < TILES) {
            unsigned g = (unsigned)(((it + 2) * TPB + tid) * 16);
            async_b128((unsigned)(uintptr_t)&lbuf[cur][0][tid], g, pbase);
            async_b128((unsigned)(uintptr_t)&lbuf[cur][1][tid], g, tbase);
        }

        const float pv[4] = {p4.x, p4.y, p4.z, p4.w};
        const float tv[4] = {t4.x, t4.y, t4.z, t4.w};
#pragma unroll
        for (int e = 0; e < 4; ++e) {
            float p = pv[e], t = tv[e];
            sumP += p;  sumT += t;
            maxP = fmaxf(maxP, p);  maxT = fmaxf(maxT, t);
            // total_map = (p-t)^2 * (mask ? t^0.002 : 1)   [mask: t > 0.1]
            float d = p - t;
            float w = (t > THRESH) ? __powf(t, 0.002f) : 1.0f;
            lossA = fmaf(d * d, w, lossA);
            // packed 6-bit histogram counters (32 elems/thread -> no overflow)
            unsigned bp = min((unsigned)(p * 10.0f), 9u);
            unsigned bt = min((unsigned)(t * 10.0f), 9u);
            packP += 1ull << (bp * 6u);
            packT += 1ull << (bt * 6u);
        }
    }

    // ---- wave32 reductions of float stats ----
#pragma unroll
    for (int off = 16; off > 0; off >>= 1) {
        lossA += __shfl_xor(lossA, off, 32);
        sumP  += __shfl_xor(sumP,  off, 32);
        sumT  += __shfl_xor(sumT,  off, 32);
        maxP = fmaxf(maxP, __shfl_xor(maxP, off, 32));
        maxT = fmaxf(maxT, __shfl_xor(maxT, off, 32));
    }

    // ---- histogram: unpack 6-bit fields, combine in LDS, then global ----
    __syncthreads();   // bhist init visible
#pragma unroll
    for (int i = 0; i < 10; ++i) {
        atomicAdd(&bhist[i],      (unsigned)((packP >> (6 * i)) & 63ull));
        atomicAdd(&bhist[10 + i], (unsigned)((packT >> (6 * i)) & 63ull));
    }

    float* L  = ws;
    float* SP = ws + 64;
    float* ST = ws + 128;
    unsigned* wsu = (unsigned*)ws;
    unsigned* MP = wsu + 192;
    unsigned* MT = wsu + 256;
    unsigned* HP = wsu + 320;
    unsigned* HT = wsu + 960;

    if ((tid & 31) == 0) {
        atomicAdd(&L[b],  lossA);
        atomicAdd(&SP[b], sumP);
        atomicAdd(&ST[b], sumT);
        atomicMax(&MP[b], __float_as_uint(maxP));
        atomicMax(&MT[b], __float_as_uint(maxT));
    }
    __syncthreads();
    if (tid < 10)                 atomicAdd(&HP[b * 10 + tid],        bhist[tid]);
    else if (tid < 20)            atomicAdd(&HT[b * 10 + (tid - 10)], bhist[tid]);
}

// ---------------------------------------------------------------------------
// Finalize: one wave. Four independent 64-way reductions done on the matrix
// pipe: V_WMMA_F32_16X16X4_F32 with B = ones sums A's 16x4 = 64 f32 values
// into row-sums; 8 accumulator VGPRs per lane cover 8 rows, one xor-16
// shuffle merges the two half-sums.
// ---------------------------------------------------------------------------
__global__ void finalize_kernel(const float* __restrict__ ws,
                                float* __restrict__ out) {
    const int l = threadIdx.x;                  // 0..31, single wave32
    const int j0 = (l & 15) * 4 + (l >> 4) * 2; // A slot (M=l%16, K=2*(l/16))

    const float*    L   = ws;
    const float*    SP  = ws + 64;
    const float*    ST  = ws + 128;
    const unsigned* wsu = (const unsigned*)ws;
    const unsigned* MP  = wsu + 192;
    const unsigned* MT  = wsu + 256;
    const unsigned* HP  = wsu + 320;
    const unsigned* HT  = wsu + 960;

    v2f aL = { L[j0], L[j0 + 1] };
    v2f aS = { fabsf(SP[j0] - ST[j0]), fabsf(SP[j0 + 1] - ST[j0 + 1]) };
    v2f aM = { fabsf(__uint_as_float(MP[j0])     - __uint_as_float(MT[j0])),
               fabsf(__uint_as_float(MP[j0 + 1]) - __uint_as_float(MT[j0 + 1])) };
    float h0 = 0.f, h1 = 0.f;
#pragma unroll
    for (int t = 0; t < 10; ++t) {              // 640 hist diffs -> 64 partials
        float d0 = (float)((int)HP[j0 + 64 * t]     - (int)HT[j0 + 64 * t]);
        float d1 = (float)((int)HP[j0 + 1 + 64 * t] - (int)HT[j0 + 1 + 64 * t]);
        h0 = fmaf(d0, d0, h0);
        h1 = fmaf(d1, d1, h1);
    }
    v2f aH = { h0, h1 };

    v2f ones = { 1.0f, 1.0f };
    v8f z = { 0.f, 0.f, 0.f, 0.f, 0.f, 0.f, 0.f, 0.f };

    v8f dL = __builtin_amdgcn_wmma_f32_16x16x4_f32(false, aL, false, ones, (short)0, z, false, false);
    v8f dS = __builtin_amdgcn_wmma_f32_16x16x4_f32(false, aS, false, ones, (short)0, z, false, false);
    v8f dM = __builtin_amdgcn_wmma_f32_16x16x4_f32(false, aM, false, ones, (short)0, z, false, false);
    v8f dH = __builtin_amdgcn_wmma_f32_16x16x4_f32(false, aH, false, ones, (short)0, z, false, false);

    float sL = 0.f, sS = 0.f, sM = 0.f, sH = 0.f;
#pragma unroll
    for (int i = 0; i < 8; ++i) { sL += dL[i]; sS += dS[i]; sM += dM[i]; sH += dH[i]; }
    sL += __shfl_xor(sL, 16, 32);
    sS += __shfl_xor(sS, 16, 32);
    sM += __shfl_xor(sM, 16, 32);
    sH += __shfl_xor(sH, 16, 32);

    if (l == 0) {
        const float numel   = 262144.0f;
        const float inv     = 1.0f / (numel + 1e-8f);
        const float histden = 1.0f / (numel * numel) / 640.0f;
        out[0] = sL + 0.001f * sM * inv + 0.001f * sS * inv + 0.001f * sH * histden;
    }
}

extern "C" void kernel_launch(void* const* d_in, const int* in_sizes, int n_in,
                              void* d_out, int out_size, void* d_ws, size_t ws_size,
                              hipStream_t stream) {
    const float* pred = (const float*)d_in[0];
    const float* targ = (const float*)d_in[1];
    float* out = (float*)d_out;
    float* ws  = (float*)d_ws;
    (void)in_sizes; (void)n_in; (void)out_size; (void)ws_size;

    hipMemsetAsync(ws, 0, WS_WORDS * sizeof(float), stream);
    stats_kernel<<<dim3(NBATCH * BPB), dim3(TPB), 0, stream>>>(pred, targ, ws);
    finalize_kernel<<<dim3(1), dim3(32), 0, stream>>>(ws, out);
}